// ConfusionMat_82832739271313
// MI455X (gfx1250) — compile-verified
//
#include <hip/hip_runtime.h>
#include <hip/hip_bf16.h>

// Problem constants (fixed by the reference harness).
#define CLASSES   19
#define BINS      (CLASSES * CLASSES)   // 361
#define BATCH     8
#define IMG_H     512
#define IMG_W     512
#define HW        (IMG_H * IMG_W)       // 262144
#define NPIX      (BATCH * HW)          // 2097152
#define TPB       256
#define NBLOCKS   1024
#define CHUNKS_PER_BLOCK (NPIX / (TPB * NBLOCKS))  // 8, exact

// ---------------------------------------------------------------------------
// CDNA5 async-to-LDS helpers (inline asm: portable across ROCm7.2 / clang-23).
// global_load_async_to_lds_b32: per-lane copy of 4 bytes from the global address in vaddr to the LDS address in vdst,
// tracked by ASYNCcnt (non-blocking, no VGPR destination).
// ---------------------------------------------------------------------------
__device__ __forceinline__ void async_ld_b32(unsigned lds_byte_addr, const float* gptr) {
    asm volatile("global_load_async_to_lds_b32 %0, %1, off"
                 :: "v"(lds_byte_addr), "v"(gptr)
                 : "memory");
}
__device__ __forceinline__ void wait_async_le19() {
    asm volatile("s_wait_asynccnt 19" ::: "memory");
}
__device__ __forceinline__ void wait_async_0() {
    asm volatile("s_wait_asynccnt 0" ::: "memory");
}

// ---------------------------------------------------------------------------
// Kernel 1: zero the 361-bin integer histogram in workspace.
// ---------------------------------------------------------------------------
__global__ void ConfusionMat_zero_kernel(unsigned int* __restrict__ hist) {
    int i = blockIdx.x * blockDim.x + threadIdx.x;
    if (i < BINS) hist[i] = 0u;
}

// ---------------------------------------------------------------------------
// Kernel 2: streaming argmax + LDS histogram.
//   - double-buffered async staging of 19 channel values per pixel into LDS
//   - per-block LDS histogram (ds_add_u32), single global flush per block
// ---------------------------------------------------------------------------
__global__ __launch_bounds__(TPB)
void ConfusionMat_main_kernel(const float* __restrict__ inp,
                              const int*   __restrict__ tgt,
                              unsigned int* __restrict__ hist) {
    __shared__ float        buf[2][CLASSES][TPB];   // 38 KB double buffer
    __shared__ unsigned int lh[BINS];               // block-local histogram

    const int tid = threadIdx.x;

    for (int i = tid; i < BINS; i += TPB) lh[i] = 0u;
    __syncthreads();

    // Low 32 bits of a generic pointer into __shared__ == LDS byte offset
    // (the high 32 bits of a generic LDS address hold the shared aperture base).
    const unsigned lds_base = (unsigned)(size_t)(&buf[0][0][0]);
    const int chunk0 = blockIdx.x * CHUNKS_PER_BLOCK;

    // Issue 19 async channel loads for one 256-pixel chunk into buffer k.
    // Lanes step w -> each channel load is a contiguous 128B/wave transfer.
    auto issue = [&](int chunk, int k) {
        const int p  = chunk * TPB + tid;        // global pixel id
        const int b  = p / HW;
        const int hw = p - b * HW;
        const float* g = inp + (size_t)(b * CLASSES) * HW + hw;
        unsigned l = lds_base + (unsigned)((k * CLASSES) * TPB + tid) * 4u;
#pragma unroll
        for (int c = 0; c < CLASSES; ++c) {
            async_ld_b32(l, g);
            g += HW;                              // next channel plane
            l += TPB * 4u;
        }
    };

    issue(chunk0, 0);                             // prologue

    for (int i = 0; i < CHUNKS_PER_BLOCK; ++i) {
        const int  cur  = i & 1;
        const bool more = (i + 1) < CHUNKS_PER_BLOCK;

        if (more) {
            issue(chunk0 + i + 1, cur ^ 1);       // prefetch next chunk
            wait_async_le19();                    // in-order retire => chunk i done
        } else {
            wait_async_0();                       // drain
        }

        // Argmax over 19 staged channel values (first-max wins, like jnp.argmax).
        float best = buf[cur][0][tid];
        int   bc   = 0;
#pragma unroll
        for (int c = 1; c < CLASSES; ++c) {
            float v = buf[cur][c][tid];
            if (v > best) { best = v; bc = c; }
        }

        const int p = (chunk0 + i) * TPB + tid;
        const int t = tgt[p];
        atomicAdd(&lh[t * CLASSES + bc], 1u);     // ds_add_u32
    }

    __syncthreads();
    for (int i = tid; i < BINS; i += TPB) {
        unsigned v = lh[i];
        if (v) atomicAdd(&hist[i], v);            // global_atomic_add_u32
    }
}

// ---------------------------------------------------------------------------
// Kernel 3: convert integer counts -> float output (counts < 2^24: exact).
// ---------------------------------------------------------------------------
__global__ void ConfusionMat_final_kernel(const unsigned int* __restrict__ hist,
                                          float* __restrict__ out) {
    int i = blockIdx.x * blockDim.x + threadIdx.x;
    if (i < BINS) out[i] = (float)hist[i];
}

// ---------------------------------------------------------------------------
extern "C" void kernel_launch(void* const* d_in, const int* in_sizes, int n_in,
                              void* d_out, int out_size, void* d_ws, size_t ws_size,
                              hipStream_t stream) {
    const float* inp  = (const float*)d_in[0];   // [8,19,512,512] f32
    const int*   tgt  = (const int*)  d_in[1];   // [8,512,512]    i32
    unsigned int* hist = (unsigned int*)d_ws;    // 361 u32 scratch
    float*       out  = (float*)d_out;           // 361 outputs

    ConfusionMat_zero_kernel<<<(BINS + TPB - 1) / TPB, TPB, 0, stream>>>(hist);
    ConfusionMat_main_kernel<<<NBLOCKS, TPB, 0, stream>>>(inp, tgt, hist);
    ConfusionMat_final_kernel<<<(BINS + TPB - 1) / TPB, TPB, 0, stream>>>(hist, out);
}